// GNNReg_42314017800206
// MI455X (gfx1250) — compile-verified
//
#include <hip/hip_runtime.h>

#define NN   50000
#define EE   1600000
#define FF   64
#define FEA  16
#define DIM  128
#define GG   2048

typedef __attribute__((ext_vector_type(2))) float v2f;
typedef __attribute__((ext_vector_type(8))) float v8f;

__device__ __forceinline__ v8f wmma4(v2f a, v2f b, v8f c) {
  // V_WMMA_F32_16X16X4_F32 : D = A(16x4) * B(4x16) + C(16x16)
  // 8-arg pattern: (neg_a, A, neg_b, B, c_mod, C, reuse_a, reuse_b)
  return __builtin_amdgcn_wmma_f32_16x16x4_f32(
      false, a, false, b, (short)0, c, false, false);
}

// ---------------------------------------------------------------------------
// zero helper (float4 grid-stride)
// ---------------------------------------------------------------------------
__global__ void zero_kernel(float* __restrict__ p, int n4) {
  int i = blockIdx.x * blockDim.x + threadIdx.x;
  int stride = gridDim.x * blockDim.x;
  float4 z; z.x = 0.f; z.y = 0.f; z.z = 0.f; z.w = 0.f;
  for (; i < n4; i += stride) ((float4*)p)[i] = z;
}

// ---------------------------------------------------------------------------
// K1: h = relu(x @ W0 + b0)      x:[N,64]  W0:[64,128]  -> h:[N,128]
// one wave per 16x16 output tile; K stepped by 4 via f32 WMMA
// ---------------------------------------------------------------------------
__global__ void lin0_kernel(const float* __restrict__ x, const float* __restrict__ W0,
                            const float* __restrict__ b0, float* __restrict__ h) {
  int wave = blockIdx.x * (blockDim.x >> 5) + (threadIdx.x >> 5);
  int lane = threadIdx.x & 31;
  int kh = lane >> 4, l = lane & 15;
  int row0 = (wave >> 3) << 4;          // 8 col tiles (DIM/16)
  int col0 = (wave & 7) << 4;
  if (row0 >= NN) return;
  v8f acc = {};
#pragma unroll
  for (int k = 0; k < FF; k += 4) {
    int kk = k + 2 * kh;
    v2f a = *(const v2f*)(x + (row0 + l) * FF + kk);
    v2f b;
    b.x = W0[kk * DIM + col0 + l];
    b.y = W0[(kk + 1) * DIM + col0 + l];
    acc = wmma4(a, b, acc);
  }
  float bias = b0[col0 + l];
#pragma unroll
  for (int i = 0; i < 8; i++) {
    float v = acc[i] + bias;
    h[(row0 + i + 8 * kh) * DIM + col0 + l] = v > 0.f ? v : 0.f;
  }
}

// ---------------------------------------------------------------------------
// K2: per 16-edge group:  ea = edge_attr @ We + be  (WMMA, K=16)
//     msg = relu(h[src] + ea);  atomicAdd(agg[dst], msg)
// ---------------------------------------------------------------------------
__global__ void edge_kernel(const float* __restrict__ eat, const float* __restrict__ We,
                            const float* __restrict__ be,
                            const int* __restrict__ src, const int* __restrict__ dst,
                            const float* __restrict__ h, float* __restrict__ agg) {
  int wave = blockIdx.x * (blockDim.x >> 5) + (threadIdx.x >> 5);
  int lane = threadIdx.x & 31;
  int kh = lane >> 4, l = lane & 15;
  int e0 = wave << 4;
  if (e0 >= EE) return;

  // A tile (16 edges x 16 edge-features), all K preloaded
  v2f a[4];
#pragma unroll
  for (int q = 0; q < 4; q++)
    a[q] = *(const v2f*)(eat + (e0 + l) * FEA + 4 * q + 2 * kh);

  // per-lane copies of the 8 row indices this lane scatters (broadcast loads)
  int sIdx[8], dIdx[8];
#pragma unroll
  for (int i = 0; i < 8; i++) {
    sIdx[i] = src[e0 + i + 8 * kh];
    dIdx[i] = dst[e0 + i + 8 * kh];
  }

#pragma unroll
  for (int ct = 0; ct < DIM / 16; ct++) {
    int col0 = ct << 4;
    v8f acc = {};
#pragma unroll
    for (int q = 0; q < 4; q++) {
      int kk = 4 * q + 2 * kh;
      v2f b;
      b.x = We[kk * DIM + col0 + l];
      b.y = We[(kk + 1) * DIM + col0 + l];
      acc = wmma4(a[q], b, acc);
    }
    float bias = be[col0 + l];
#pragma unroll
    for (int i = 0; i < 8; i++) {
      float v = acc[i] + bias + h[sIdx[i] * DIM + col0 + l];
      v = v > 0.f ? v : 0.f;
      atomicAdd(agg + dIdx[i] * DIM + col0 + l, v);
    }
  }
}

// ---------------------------------------------------------------------------
// K3a: t = relu((h + agg) @ W1 + b1)   [N,128] -> [N,256]
// ---------------------------------------------------------------------------
__global__ void mlp1_kernel(const float* __restrict__ h, const float* __restrict__ agg,
                            const float* __restrict__ W1, const float* __restrict__ b1,
                            float* __restrict__ t) {
  int wave = blockIdx.x * (blockDim.x >> 5) + (threadIdx.x >> 5);
  int lane = threadIdx.x & 31;
  int kh = lane >> 4, l = lane & 15;
  int row0 = (wave >> 4) << 4;          // 16 col tiles (2*DIM/16)
  int col0 = (wave & 15) << 4;
  if (row0 >= NN) return;
  v8f acc = {};
#pragma unroll 8
  for (int k = 0; k < DIM; k += 4) {
    int kk = k + 2 * kh;
    v2f a  = *(const v2f*)(h   + (row0 + l) * DIM + kk);
    v2f a2 = *(const v2f*)(agg + (row0 + l) * DIM + kk);
    a.x += a2.x; a.y += a2.y;
    v2f b;
    b.x = W1[kk * (2 * DIM) + col0 + l];
    b.y = W1[(kk + 1) * (2 * DIM) + col0 + l];
    acc = wmma4(a, b, acc);
  }
  float bias = b1[col0 + l];
#pragma unroll
  for (int i = 0; i < 8; i++) {
    float v = acc[i] + bias;
    t[(row0 + i + 8 * kh) * (2 * DIM) + col0 + l] = v > 0.f ? v : 0.f;
  }
}

// ---------------------------------------------------------------------------
// K3b: z = relu(t @ W2 + b2); pooling: atomicAdd(g[batch[row]], z)
// ---------------------------------------------------------------------------
__global__ void mlp2_pool_kernel(const float* __restrict__ t, const float* __restrict__ W2,
                                 const float* __restrict__ b2, const int* __restrict__ batch,
                                 float* __restrict__ g) {
  int wave = blockIdx.x * (blockDim.x >> 5) + (threadIdx.x >> 5);
  int lane = threadIdx.x & 31;
  int kh = lane >> 4, l = lane & 15;
  int row0 = (wave >> 3) << 4;          // 8 col tiles
  int col0 = (wave & 7) << 4;
  if (row0 >= NN) return;
  v8f acc = {};
#pragma unroll 8
  for (int k = 0; k < 2 * DIM; k += 4) {
    int kk = k + 2 * kh;
    v2f a = *(const v2f*)(t + (row0 + l) * (2 * DIM) + kk);
    v2f b;
    b.x = W2[kk * DIM + col0 + l];
    b.y = W2[(kk + 1) * DIM + col0 + l];
    acc = wmma4(a, b, acc);
  }
  float bias = b2[col0 + l];
#pragma unroll
  for (int i = 0; i < 8; i++) {
    int m = row0 + i + 8 * kh;
    float v = acc[i] + bias;
    v = v > 0.f ? v : 0.f;
    atomicAdd(g + batch[m] * DIM + col0 + l, v);
  }
}

// ---------------------------------------------------------------------------
// K4a: r1 = relu([g | 10*temp] @ Wf1 + bf1)   (extra K=128 term folded in)
// ---------------------------------------------------------------------------
__global__ void ro1_kernel(const float* __restrict__ g, const float* __restrict__ temp,
                           const float* __restrict__ Wf1, const float* __restrict__ bf1,
                           float* __restrict__ r1) {
  int wave = blockIdx.x * (blockDim.x >> 5) + (threadIdx.x >> 5);
  int lane = threadIdx.x & 31;
  int kh = lane >> 4, l = lane & 15;
  int row0 = (wave >> 3) << 4;
  int col0 = (wave & 7) << 4;
  if (row0 >= GG) return;
  v8f acc = {};
#pragma unroll 8
  for (int k = 0; k < DIM; k += 4) {
    int kk = k + 2 * kh;
    v2f a = *(const v2f*)(g + (row0 + l) * DIM + kk);
    v2f b;
    b.x = Wf1[kk * DIM + col0 + l];
    b.y = Wf1[(kk + 1) * DIM + col0 + l];
    acc = wmma4(a, b, acc);
  }
  float wlast = Wf1[DIM * DIM + col0 + l];   // row k = DIM (temp column)
  float bias  = bf1[col0 + l];
#pragma unroll
  for (int i = 0; i < 8; i++) {
    int m = row0 + i + 8 * kh;
    float v = acc[i] + 10.f * temp[m] * wlast + bias;
    r1[m * DIM + col0 + l] = v > 0.f ? v : 0.f;
  }
}

// ---------------------------------------------------------------------------
// K4b: r2 = relu(r1 @ Wf2 + bf2)
// ---------------------------------------------------------------------------
__global__ void ro2_kernel(const float* __restrict__ r1, const float* __restrict__ Wf2,
                           const float* __restrict__ bf2, float* __restrict__ r2) {
  int wave = blockIdx.x * (blockDim.x >> 5) + (threadIdx.x >> 5);
  int lane = threadIdx.x & 31;
  int kh = lane >> 4, l = lane & 15;
  int row0 = (wave >> 3) << 4;
  int col0 = (wave & 7) << 4;
  if (row0 >= GG) return;
  v8f acc = {};
#pragma unroll 8
  for (int k = 0; k < DIM; k += 4) {
    int kk = k + 2 * kh;
    v2f a = *(const v2f*)(r1 + (row0 + l) * DIM + kk);
    v2f b;
    b.x = Wf2[kk * DIM + col0 + l];
    b.y = Wf2[(kk + 1) * DIM + col0 + l];
    acc = wmma4(a, b, acc);
  }
  float bias = bf2[col0 + l];
#pragma unroll
  for (int i = 0; i < 8; i++) {
    float v = acc[i] + bias;
    r2[(row0 + i + 8 * kh) * DIM + col0 + l] = v > 0.f ? v : 0.f;
  }
}

// ---------------------------------------------------------------------------
// K4c: out = r2 @ Wf3 + bf3   ([2048,128] x [128,1]) — wave32 reduction
// ---------------------------------------------------------------------------
__global__ void ro3_kernel(const float* __restrict__ r2, const float* __restrict__ Wf3,
                           const float* __restrict__ bf3, float* __restrict__ out) {
  int wave = blockIdx.x * (blockDim.x >> 5) + (threadIdx.x >> 5);
  int lane = threadIdx.x & 31;
  if (wave >= GG) return;
  float s = 0.f;
#pragma unroll
  for (int k = lane; k < DIM; k += 32) s += r2[wave * DIM + k] * Wf3[k];
#pragma unroll
  for (int off = 16; off > 0; off >>= 1) s += __shfl_xor(s, off, 32);
  if (lane == 0) out[wave] = s + bf3[0];
}

// ---------------------------------------------------------------------------
extern "C" void kernel_launch(void* const* d_in, const int* in_sizes, int n_in,
                              void* d_out, int out_size, void* d_ws, size_t ws_size,
                              hipStream_t stream) {
  const float* x    = (const float*)d_in[0];
  const float* eat  = (const float*)d_in[1];
  const float* temp = (const float*)d_in[2];
  const int*   eidx = (const int*)  d_in[3];
  const int*   batch= (const int*)  d_in[4];
  const float* W0   = (const float*)d_in[5];
  const float* b0   = (const float*)d_in[6];
  const float* We   = (const float*)d_in[7];
  const float* be   = (const float*)d_in[8];
  const float* W1   = (const float*)d_in[9];
  const float* b1   = (const float*)d_in[10];
  const float* W2   = (const float*)d_in[11];
  const float* b2   = (const float*)d_in[12];
  const float* Wf1  = (const float*)d_in[13];
  const float* bf1  = (const float*)d_in[14];
  const float* Wf2  = (const float*)d_in[15];
  const float* bf2  = (const float*)d_in[16];
  const float* Wf3  = (const float*)d_in[17];
  const float* bf3  = (const float*)d_in[18];
  float* out = (float*)d_out;

  const int* src = eidx;        // edge_index[0]
  const int* dst = eidx + EE;   // edge_index[1]

  float* ws   = (float*)d_ws;
  float* h    = ws;                        // [N,128]
  float* agg  = ws + (size_t)NN * DIM;     // [N,128]
  float* tbuf = ws + (size_t)2 * NN * DIM; // [N,256]
  // after mlp1, h/agg regions are dead -> reuse for readout buffers
  float* g  = ws;                          // [G,128]
  float* r1 = ws + (size_t)GG * DIM;       // [G,128]
  float* r2 = ws + (size_t)2 * GG * DIM;   // [G,128]

  const int B = 256;                       // 8 waves per block

  zero_kernel     <<<2048, B, 0, stream>>>(agg, NN * DIM / 4);
  lin0_kernel     <<<(NN / 16) * 8  / 8, B, 0, stream>>>(x, W0, b0, h);
  edge_kernel     <<<(EE / 16)      / 8, B, 0, stream>>>(eat, We, be, src, dst, h, agg);
  mlp1_kernel     <<<(NN / 16) * 16 / 8, B, 0, stream>>>(h, agg, W1, b1, tbuf);
  zero_kernel     <<<64,   B, 0, stream>>>(g, GG * DIM / 4);
  mlp2_pool_kernel<<<(NN / 16) * 8  / 8, B, 0, stream>>>(tbuf, W2, b2, batch, g);
  ro1_kernel      <<<(GG / 16) * 8  / 8, B, 0, stream>>>(g, temp, Wf1, bf1, r1);
  ro2_kernel      <<<(GG / 16) * 8  / 8, B, 0, stream>>>(r1, Wf2, bf2, r2);
  ro3_kernel      <<<GG / 8,            B, 0, stream>>>(r2, Wf3, bf3, out);
}